// GAT1Layer_77945066488525
// MI455X (gfx1250) — compile-verified
//
#include <hip/hip_runtime.h>
#include <hip/hip_bf16.h>

typedef __attribute__((ext_vector_type(2))) float v2f;
typedef __attribute__((ext_vector_type(8))) float v8f;

#define IN_F   128   // input features (= K)
#define HOUT   128   // H * OUT  (= GEMM N-dim)
#define NHEAD  2
#define CHEAD  64
#define NEG_SLOPE 0.2f

// ---------- order-preserving float<->int encoding for atomic max ----------
__device__ __forceinline__ int enc_f(float f) {
    int i = __float_as_int(f);
    return (i >= 0) ? i : (i ^ 0x7fffffff);
}
__device__ __forceinline__ float dec_f(int i) {
    return __int_as_float((i >= 0) ? i : (i ^ 0x7fffffff));
}

// ---------- init: zero d_out, zero denom, set e_max to -inf-encoding ----------
__global__ void gat_init(float* __restrict__ out, float* __restrict__ denom,
                         int* __restrict__ e_max, int n_out, int n_nh) {
    int i = blockIdx.x * blockDim.x + threadIdx.x;
    if (i < n_out) out[i] = 0.0f;
    if (i < n_nh) { denom[i] = 0.0f; e_max[i] = (int)0x80000000; }
}

// ---------- h = x @ W via V_WMMA_F32_16X16X4_F32, one wave per 16x16 tile ----------
__global__ void gat_gemm_wmma(const float* __restrict__ x, const float* __restrict__ W,
                              float* __restrict__ h, int N) {
    const int wave = threadIdx.x >> 5;           // 8 waves / block
    const int lane = threadIdx.x & 31;
    const int tilesN = HOUT / 16;                // 8
    const int mTiles = (N + 15) / 16;
    int tile = blockIdx.x * 8 + wave;
    if (tile >= mTiles * tilesN) return;
    const int tm = tile / tilesN;
    const int tn = tile % tilesN;

    const int hs = lane >> 4;                    // half select (0: K+0/1, 1: K+2/3)
    const int l  = lane & 15;
    int rowA = tm * 16 + l;                      // A: lanes 0-15 and 16-31 both cover M=0..15
    if (rowA >= N) rowA = N - 1;                 // clamp (N%16==0 in practice)
    const int colB = tn * 16 + l;

    v8f acc = {};
    const float* xrow = x + (long)rowA * IN_F;
    #pragma unroll
    for (int kk = 0; kk < IN_F; kk += 4) {
        const int kb = kk + 2 * hs;
        // A fragment: VGPR0 = K=kb, VGPR1 = K=kb+1 (16x4 fp32 layout)
        v2f a;
        a.x = xrow[kb + 0];
        a.y = xrow[kb + 1];
        // B fragment (4x16): mirrored layout, rows kb / kb+1 of W
        v2f b;
        b.x = W[(kb + 0) * HOUT + colB];
        b.y = W[(kb + 1) * HOUT + colB];
        acc = __builtin_amdgcn_wmma_f32_16x16x4_f32(
            /*neg_a=*/false, a, /*neg_b=*/false, b,
            /*c_mod=*/(short)0, acc, /*reuse_a=*/false, /*reuse_b=*/false);
    }
    // C/D layout: VGPR r -> row r (lanes 0-15) or r+8 (lanes 16-31), col = lane&15
    #pragma unroll
    for (int r = 0; r < 8; r++) {
        int row = tm * 16 + r + 8 * hs;
        if (row < N) h[(long)row * HOUT + tn * 16 + l] = acc[r];
    }
}

// ---------- per-node attention logits ----------
__global__ void gat_logits(const float* __restrict__ h,
                           const float* __restrict__ att_src, const float* __restrict__ att_dst,
                           float* __restrict__ a_src, float* __restrict__ a_dst, int N) {
    int idx = blockIdx.x * blockDim.x + threadIdx.x;   // node*NHEAD + head
    if (idx >= N * NHEAD) return;
    const int node = idx >> 1, hd = idx & 1;
    const float* hp = h + (long)node * HOUT + hd * CHEAD;
    const float* as = att_src + hd * CHEAD;
    const float* ad = att_dst + hd * CHEAD;
    float s = 0.f, d = 0.f;
    #pragma unroll 8
    for (int c = 0; c < CHEAD; c++) { float v = hp[c]; s += v * as[c]; d += v * ad[c]; }
    a_src[idx] = s;
    a_dst[idx] = d;
}

__device__ __forceinline__ void edge_sd(const long long* src, const long long* dst,
                                        int e, int E, int& s, int& d) {
    if (e < E) { s = (int)src[e]; d = (int)dst[e]; }
    else       { s = d = e - E; }                     // self loop
}

__device__ __forceinline__ float leaky(float v) { return v > 0.f ? v : NEG_SLOPE * v; }

// ---------- pass A: segment max of leaky-relu logits ----------
__global__ void gat_edge_max(const long long* __restrict__ src, const long long* __restrict__ dst,
                             int E, int N,
                             const float* __restrict__ a_src, const float* __restrict__ a_dst,
                             int* __restrict__ e_max) {
    int e = blockIdx.x * blockDim.x + threadIdx.x;
    if (e >= E + N) return;
    int s, d; edge_sd(src, dst, e, E, s, d);
    #pragma unroll
    for (int hd = 0; hd < NHEAD; hd++) {
        float v = leaky(a_src[s * NHEAD + hd] + a_dst[d * NHEAD + hd]);
        atomicMax(&e_max[d * NHEAD + hd], enc_f(v));
    }
}

// ---------- pass B: segment sum of exp(e - max) ----------
__global__ void gat_edge_sum(const long long* __restrict__ src, const long long* __restrict__ dst,
                             int E, int N,
                             const float* __restrict__ a_src, const float* __restrict__ a_dst,
                             const int* __restrict__ e_max, float* __restrict__ denom) {
    int e = blockIdx.x * blockDim.x + threadIdx.x;
    if (e >= E + N) return;
    int s, d; edge_sd(src, dst, e, E, s, d);
    #pragma unroll
    for (int hd = 0; hd < NHEAD; hd++) {
        float v = leaky(a_src[s * NHEAD + hd] + a_dst[d * NHEAD + hd]);
        float m = dec_f(e_max[d * NHEAD + hd]);
        atomicAdd(&denom[d * NHEAD + hd], expf(v - m));
    }
}

// ---------- pass C: weighted scatter-aggregate, head mean folded in ----------
// one wave per edge; lane handles output channels {lane, lane+32}
__global__ void gat_edge_agg(const long long* __restrict__ src, const long long* __restrict__ dst,
                             int E, int N,
                             const float* __restrict__ a_src, const float* __restrict__ a_dst,
                             const int* __restrict__ e_max, const float* __restrict__ denom,
                             const float* __restrict__ h, float* __restrict__ out) {
    int wid  = (blockIdx.x * blockDim.x + threadIdx.x) >> 5;
    int lane = threadIdx.x & 31;
    if (wid >= E + N) return;
    int s, d; edge_sd(src, dst, wid, E, s, d);
    float alpha[NHEAD];
    #pragma unroll
    for (int hd = 0; hd < NHEAD; hd++) {
        float v = leaky(a_src[s * NHEAD + hd] + a_dst[d * NHEAD + hd]);
        float m = dec_f(e_max[d * NHEAD + hd]);
        alpha[hd] = expf(v - m) / denom[d * NHEAD + hd];
    }
    const float* hp = h + (long)s * HOUT;
    #pragma unroll
    for (int t = 0; t < 2; t++) {
        int c = lane + 32 * t;
        float val = 0.5f * (alpha[0] * hp[c] + alpha[1] * hp[CHEAD + c]);
        atomicAdd(&out[(long)d * CHEAD + c], val);
    }
}

// ---------- finalize: bias + relu in place ----------
__global__ void gat_finalize(float* __restrict__ out, const float* __restrict__ bias, int n_out) {
    int i = blockIdx.x * blockDim.x + threadIdx.x;
    if (i >= n_out) return;
    float v = out[i] + bias[i & (CHEAD - 1)];
    out[i] = v > 0.f ? v : 0.f;
}

extern "C" void kernel_launch(void* const* d_in, const int* in_sizes, int n_in,
                              void* d_out, int out_size, void* d_ws, size_t ws_size,
                              hipStream_t stream) {
    const float*      x        = (const float*)d_in[0];
    const long long*  eidx     = (const long long*)d_in[1];
    const float*      W        = (const float*)d_in[2];
    const float*      att_src  = (const float*)d_in[3];
    const float*      att_dst  = (const float*)d_in[4];
    const float*      bias     = (const float*)d_in[5];
    float* out = (float*)d_out;

    const int N  = in_sizes[0] / IN_F;       // 50000
    const int E  = in_sizes[1] / 2;          // 1600000
    const int ET = E + N;                    // with self loops
    const long long* src = eidx;
    const long long* dst = eidx + E;

    // workspace layout
    float* ws      = (float*)d_ws;
    float* h       = ws;                           // N*128
    float* a_src   = h + (long)N * HOUT;           // N*2
    float* a_dst   = a_src + N * NHEAD;            // N*2
    float* denom   = a_dst + N * NHEAD;            // N*2
    int*   e_max   = (int*)(denom + N * NHEAD);    // N*2

    const int BLK = 256;
    const int n_out = N * CHEAD;
    const int n_nh  = N * NHEAD;

    gat_init<<<(n_out + BLK - 1) / BLK, BLK, 0, stream>>>(out, denom, e_max, n_out, n_nh);

    const int tiles = ((N + 15) / 16) * (HOUT / 16);
    gat_gemm_wmma<<<(tiles + 7) / 8, BLK, 0, stream>>>(x, W, h, N);

    gat_logits<<<(n_nh + BLK - 1) / BLK, BLK, 0, stream>>>(h, att_src, att_dst, a_src, a_dst, N);

    gat_edge_max<<<(ET + BLK - 1) / BLK, BLK, 0, stream>>>(src, dst, E, N, a_src, a_dst, e_max);
    gat_edge_sum<<<(ET + BLK - 1) / BLK, BLK, 0, stream>>>(src, dst, E, N, a_src, a_dst, e_max, denom);

    // one wave (32 lanes) per edge -> 8 edges per 256-thread block
    gat_edge_agg<<<(ET + 7) / 8, BLK, 0, stream>>>(src, dst, E, N, a_src, a_dst, e_max, denom, h, out);

    gat_finalize<<<(n_out + BLK - 1) / BLK, BLK, 0, stream>>>(out, bias, n_out);
}